// GridEncoder_86569360818383
// MI455X (gfx1250) — compile-verified
//
#include <hip/hip_runtime.h>
#include <hip/hip_bf16.h>
#include <stdint.h>

// Instant-NGP grid encode, MI455X (gfx1250, wave32).
// Level-table constants baked at compile time (L=16, H=16, S=1, log2_hashmap=19).
// Levels 0..2: dense indexing. Levels 3..15: fast_hash, hsize = 2^19 -> & 0x7FFFF.
__constant__ uint32_t kOff[16] = {
    0u,       4920u,    40864u,   315496u,  839784u,  1364072u,
    1888360u, 2412648u, 2936936u, 3461224u, 3985512u, 4509800u,
    5034088u, 5558376u, 6082664u, 6606952u};

#define HASH_P1 2654435761u
#define HASH_P2 805459861u

// Block = 32 points (threadIdx.x = lane) x 16 levels (threadIdx.y = wave).
// Level is wave-uniform on wave32 -> no divergence on dense/hash branch.
__global__ __launch_bounds__(512) void grid_encode_kernel(
    const float* __restrict__ in, const float* __restrict__ emb,
    float* __restrict__ out, int n) {
  __shared__ float s_in[96];        // 32 points x 3 coords, loaded once per block
  __shared__ float s_out[32 * 34];  // 32 points x 32 outputs, pad 34 -> conflict-free

  const int lane = threadIdx.x;  // point within tile
  const int lvl  = threadIdx.y;  // level
  const int tid  = lvl * 32 + lane;
  const int64_t pbase = (int64_t)blockIdx.x * 32;

  // Cooperative input staging via the CDNA5 async memory->LDS path:
  // one global_load_async_to_lds_b32 per lane (waves 0..2), no VGPR bounce.
  // Out-of-range lanes zero-fill via ds_store at disjoint addresses (no race).
  if (tid < 96) {
    const int64_t gi = pbase * 3 + tid;
    if (gi < (int64_t)n * 3) {
      const float* gp = in + gi;
      unsigned loff =
          (unsigned)(size_t)(__attribute__((address_space(3))) float*)&s_in[tid];
      asm volatile("global_load_async_to_lds_b32 %0, %1, off"
                   :: "v"(loff), "v"(gp) : "memory");
    } else {
      s_in[tid] = 0.0f;
    }
  }
  asm volatile("s_wait_asynccnt 0x0" ::: "memory");
  __syncthreads();

  // pos = ((x+1)*0.5) * scale + 0.5, with unfused rounding to match reference
  // (scale = 16<<lvl - 1 is exact in fp32 up to 524287).
  const float scale = (float)((16u << lvl) - 1u);
  float f[3];
  uint32_t pg[3];
#pragma unroll
  for (int d = 0; d < 3; ++d) {
    float x   = __fmul_rn(__fadd_rn(s_in[lane * 3 + d], 1.0f), 0.5f);
    float pos = __fadd_rn(__fmul_rn(x, scale), 0.5f);
    float fl  = floorf(pos);
    f[d]  = pos - fl;  // exact
    pg[d] = (uint32_t)fl;
  }
  const float w0x = 1.0f - f[0], w0y = 1.0f - f[1], w0z = 1.0f - f[2];

  // Per-axis index contributions for the two corner choices on each axis.
  uint32_t ix[2], iy[2], iz[2];
  if (lvl < 3) {  // dense: idx = x + y*r + z*r^2, r = (16<<lvl)+1
    const uint32_t r = (16u << lvl) + 1u;
    ix[0] = pg[0];          ix[1] = pg[0] + 1u;
    iy[0] = pg[1] * r;      iy[1] = (pg[1] + 1u) * r;
    iz[0] = pg[2] * r * r;  iz[1] = (pg[2] + 1u) * r * r;
  } else {        // hashed: idx = (x ^ y*P1 ^ z*P2) & (2^19-1)
    ix[0] = pg[0];           ix[1] = pg[0] + 1u;
    iy[0] = pg[1] * HASH_P1; iy[1] = (pg[1] + 1u) * HASH_P1;
    iz[0] = pg[2] * HASH_P2; iz[1] = (pg[2] + 1u) * HASH_P2;
  }
  const uint32_t off = kOff[lvl];
  const float2* __restrict__ e2 = (const float2*)emb;

  // Compute all 8 indices, then issue all 8 b64 gathers back-to-back
  // (8 loads in flight per thread; table is L2-resident: 57 MB << 192 MB).
  uint32_t idx[8];
#pragma unroll
  for (int c = 0; c < 8; ++c) {
    const uint32_t bx = c & 1, by = (c >> 1) & 1, bz = (c >> 2) & 1;
    uint32_t id;
    if (lvl < 3) id = ix[bx] + iy[by] + iz[bz];
    else         id = (ix[bx] ^ iy[by] ^ iz[bz]) & 0x7FFFFu;
    idx[c] = id + off;
  }
  float2 v[8];
#pragma unroll
  for (int c = 0; c < 8; ++c) v[c] = e2[idx[c]];

  // Accumulate in the reference's corner order and weight association.
  float o0 = 0.0f, o1 = 0.0f;
#pragma unroll
  for (int c = 0; c < 8; ++c) {
    const int bx = c & 1, by = (c >> 1) & 1, bz = (c >> 2) & 1;
    const float w = ((bx ? f[0] : w0x) * (by ? f[1] : w0y)) * (bz ? f[2] : w0z);
    o0 += w * v[c].x;
    o1 += w * v[c].y;
  }

  // Stage [point][level*2+c] in LDS (stride 34: gcd(34,64)=2 -> lanes hit
  // 32 distinct even banks, conflict-free b64 stores on wave32).
  s_out[lane * 34 + lvl * 2 + 0] = o0;
  s_out[lane * 34 + lvl * 2 + 1] = o1;
  __syncthreads();

  // Coalesced streaming store of the 4 KB tile. Nontemporal: keep the 64 MB
  // output stream from evicting the L2-resident embedding table.
#pragma unroll
  for (int k = 0; k < 2; ++k) {
    const int i = tid + k * 512;
    const int p = i >> 5, c = i & 31;
    const int64_t g = pbase + p;
    if (g < n)
      __builtin_nontemporal_store(s_out[p * 34 + c], &out[g * 32 + c]);
  }
}

extern "C" void kernel_launch(void* const* d_in, const int* in_sizes, int n_in,
                              void* d_out, int out_size, void* d_ws, size_t ws_size,
                              hipStream_t stream) {
  const float* inputs = (const float*)d_in[0];  // [B, 3] fp32
  const float* emb    = (const float*)d_in[1];  // [7131240, 2] fp32
  float* out          = (float*)d_out;          // [B, 32] fp32
  const int n = in_sizes[0] / 3;
  const int grid = (n + 31) / 32;
  dim3 block(32, 16);
  grid_encode_kernel<<<grid, block, 0, stream>>>(inputs, emb, out, n);
}